// SimGCL_Encoder_16724602651084
// MI455X (gfx1250) — compile-verified
//
#include <hip/hip_runtime.h>
#include <cstdint>

#define USER_NUM 100000
#define ITEM_NUM 200000
#define NTOT     (USER_NUM + ITEM_NUM)   // 300000
#define EMB      64
#define N_LAYERS 3
#define NNZ_EDGES 9600000
#define EPS_F    0.1f

typedef uint32_t u32x4 __attribute__((ext_vector_type(4)));
typedef int      i32x8 __attribute__((ext_vector_type(8)));
typedef int      i32x4 __attribute__((ext_vector_type(4)));

#define TILE    1024                 // edges staged per TDM tile
#define NITER   8                    // tiles per workgroup
#define EPB     (TILE * NITER)       // 8192 edges per block
#define THREADS 256                  // 8 wave32 waves
#define WAVES   (THREADS / 32)
#define EPW     (TILE / WAVES)       // 128 edges per wave per tile

// Low 32 bits of a flat shared-memory pointer == LDS byte offset
// (flat->LDS mapping truncates to addr[31:0], relative to the WG's LDS base).
__device__ __forceinline__ uint32_t lds_offset(const void* p) {
  return (uint32_t)(uintptr_t)p;
}

// Build a 1-D TDM descriptor and issue TENSOR_LOAD_TO_LDS.
// valid_elems clamps tensor_dim0 so tail reads are hardware zero-filled
// (and never touch memory past the allocation).
__device__ __forceinline__ void tdm_load_1d(uint32_t lds_off, const void* gptr,
                                            uint32_t valid_elems, uint32_t tile_elems) {
  uint64_t ga = (uint64_t)(uintptr_t)gptr;
  u32x4 g0;
  g0[0] = 1u;                                                  // count=1, user desc
  g0[1] = lds_off;                                             // lds_addr
  g0[2] = (uint32_t)ga;                                        // global_addr[31:0]
  g0[3] = ((uint32_t)(ga >> 32) & 0x01FFFFFFu) | (2u << 30);   // addr[56:32] | type=2
  i32x8 g1;
  g1[0] = (int)(2u << 16);                                     // data_size=2 -> 4 bytes
  g1[1] = (int)((valid_elems & 0xFFFFu) << 16);                // tensor_dim0[15:0]
  g1[2] = (int)(((valid_elems >> 16) & 0xFFFFu) | (1u << 16)); // tensor_dim0[31:16], tensor_dim1=1
  g1[3] = (int)(tile_elems << 16);                             // tile_dim0
  g1[4] = 0;                                                   // tile_dim1=0, tile_dim2=0 (1-D)
  g1[5] = (int)tile_elems;                                     // tensor_dim0_stride (unused for 1-D)
  g1[6] = 0;
  g1[7] = 0;
  i32x4 gz4; gz4[0] = 0; gz4[1] = 0; gz4[2] = 0; gz4[3] = 0;   // groups 2/3: <=2-D tensor
  i32x8 gz8;
  gz8[0] = 0; gz8[1] = 0; gz8[2] = 0; gz8[3] = 0;
  gz8[4] = 0; gz8[5] = 0; gz8[6] = 0; gz8[7] = 0;
  // amdgpu-toolchain (clang-23) 6-arg form:
  // (uint32x4 g0, int32x8 g1, int32x4, int32x4, int32x8, i32 cpol)
  __builtin_amdgcn_tensor_load_to_lds(g0, g1, gz4, gz4, gz8, 0);
}

__global__ void __launch_bounds__(THREADS)
init_kernel(const float* __restrict__ user, const float* __restrict__ item,
            float* __restrict__ ego, float* __restrict__ nxt, float* __restrict__ acc) {
  size_t i = (size_t)blockIdx.x * blockDim.x + threadIdx.x;
  const size_t total = (size_t)NTOT * EMB;
  if (i >= total) return;
  const size_t split = (size_t)USER_NUM * EMB;
  float v = (i < split) ? user[i] : item[i - split];
  ego[i] = v;
  nxt[i] = 0.0f;
  acc[i] = 0.0f;
}

__global__ void __launch_bounds__(THREADS)
scatter_kernel(const int* __restrict__ rows, const int* __restrict__ cols,
               const float* __restrict__ vals, const float* __restrict__ ego,
               float* __restrict__ nxt) {
  __shared__ int   s_rows[2][TILE];
  __shared__ int   s_cols[2][TILE];
  __shared__ float s_vals[2][TILE];

  const int lane = threadIdx.x & 31;
  const int wid  = threadIdx.x >> 5;
  const long long blockBase = (long long)blockIdx.x * EPB;
  const bool issuer = (wid == 0);

  // Prologue: stage tile 0 into buffer 0.
  if (issuer) {
    if (blockBase < (long long)NNZ_EDGES) {
      long long rem = (long long)NNZ_EDGES - blockBase;
      uint32_t valid = (rem < (long long)TILE) ? (uint32_t)rem : (uint32_t)TILE;
      tdm_load_1d(lds_offset(&s_rows[0][0]), rows + blockBase, valid, TILE);
      tdm_load_1d(lds_offset(&s_cols[0][0]), cols + blockBase, valid, TILE);
      tdm_load_1d(lds_offset(&s_vals[0][0]), vals + blockBase, valid, TILE);
    }
  }

  for (int it = 0; it < NITER; ++it) {
    const int buf = it & 1;
    const long long tileBase = blockBase + (long long)it * TILE;

    if (issuer) {
      const long long nbase = tileBase + TILE;
      const bool hasNext = (it + 1 < NITER) && (nbase < (long long)NNZ_EDGES);
      if (hasNext) {
        long long rem = (long long)NNZ_EDGES - nbase;
        uint32_t valid = (rem < (long long)TILE) ? (uint32_t)rem : (uint32_t)TILE;
        const int nb = buf ^ 1;
        tdm_load_1d(lds_offset(&s_rows[nb][0]), rows + nbase, valid, TILE);
        tdm_load_1d(lds_offset(&s_cols[nb][0]), cols + nbase, valid, TILE);
        tdm_load_1d(lds_offset(&s_vals[nb][0]), vals + nbase, valid, TILE);
        __builtin_amdgcn_s_wait_tensorcnt(3);   // 3 new pending; current tile's 3 done
      } else {
        __builtin_amdgcn_s_wait_tensorcnt(0);
      }
    }
    __syncthreads();   // current tile visible in LDS to all waves

    if (tileBase < (long long)NNZ_EDGES) {
      const int eBeg = wid * EPW;
      #pragma unroll 4
      for (int e = 0; e < EPW; ++e) {
        const long long g = tileBase + eBeg + e;
        if (g >= (long long)NNZ_EDGES) break;
        // Uniform-address LDS reads -> hardware broadcast to all 32 lanes.
        const int   r = s_rows[buf][eBeg + e];
        const int   c = s_cols[buf][eBeg + e];
        const float v = s_vals[buf][eBeg + e];
        // Lane l owns embedding elements [2l, 2l+1]: coalesced 256B row gather.
        const float2 src = *(const float2*)(ego + ((size_t)c << 6) + (size_t)(lane << 1));
        float* dst = nxt + ((size_t)r << 6) + (size_t)(lane << 1);
        __hip_atomic_fetch_add(dst + 0, v * src.x, __ATOMIC_RELAXED, __HIP_MEMORY_SCOPE_AGENT);
        __hip_atomic_fetch_add(dst + 1, v * src.y, __ATOMIC_RELAXED, __HIP_MEMORY_SCOPE_AGENT);
      }
    }
    __syncthreads();   // buffer free before it is overwritten at it+2
  }
}

__global__ void __launch_bounds__(THREADS)
post_kernel(float* __restrict__ ego, float* __restrict__ nxt, float* __restrict__ acc,
            const float* __restrict__ noise_k, float finalScale /* 0 => not final */) {
  const int lane = threadIdx.x & 31;
  const int wid  = threadIdx.x >> 5;
  const int row  = blockIdx.x * (blockDim.x >> 5) + wid;
  if (row >= NTOT) return;
  const size_t base = ((size_t)row << 6) + (size_t)(lane << 1);

  float2 v  = *(const float2*)(nxt + base);
  float2 nz = *(const float2*)(noise_k + base);

  // ||noise_row||: wave32 butterfly reduction over 32 lanes x 2 elems.
  float ss = nz.x * nz.x + nz.y * nz.y;
  #pragma unroll
  for (int off = 16; off > 0; off >>= 1) ss += __shfl_xor(ss, off, 32);
  const float inv = EPS_F / fmaxf(sqrtf(ss), 1e-12f);

  const float sx = (v.x > 0.0f) ? 1.0f : ((v.x < 0.0f) ? -1.0f : 0.0f);
  const float sy = (v.y > 0.0f) ? 1.0f : ((v.y < 0.0f) ? -1.0f : 0.0f);
  float2 e;
  e.x = v.x + sx * nz.x * inv;
  e.y = v.y + sy * nz.y * inv;

  *(float2*)(ego + base) = e;                       // next layer's input
  *(float2*)(nxt + base) = make_float2(0.f, 0.f);   // pre-zero next layer's accumulator

  float2 a = *(const float2*)(acc + base);
  a.x += e.x; a.y += e.y;
  if (finalScale > 0.0f) { a.x *= finalScale; a.y *= finalScale; }
  *(float2*)(acc + base) = a;
}

extern "C" void kernel_launch(void* const* d_in, const int* in_sizes, int n_in,
                              void* d_out, int out_size, void* d_ws, size_t ws_size,
                              hipStream_t stream) {
  const float* user  = (const float*)d_in[0];
  const float* item  = (const float*)d_in[1];
  const float* vals  = (const float*)d_in[2];
  const float* noise = (const float*)d_in[3];
  const int*   rows  = (const int*)d_in[4];
  const int*   cols  = (const int*)d_in[5];

  float* acc = (float*)d_out;                    // acc lives in d_out (user||item order)
  float* ego = (float*)d_ws;                     // N*64 f32
  float* nxt = ego + (size_t)NTOT * EMB;         // N*64 f32

  const size_t total = (size_t)NTOT * EMB;
  init_kernel<<<(unsigned)((total + THREADS - 1) / THREADS), THREADS, 0, stream>>>(
      user, item, ego, nxt, acc);

  const int scatterGrid = (NNZ_EDGES + EPB - 1) / EPB;        // 1172
  const int postGrid    = (NTOT + WAVES - 1) / WAVES;         // 37500
  for (int k = 0; k < N_LAYERS; ++k) {
    scatter_kernel<<<scatterGrid, THREADS, 0, stream>>>(rows, cols, vals, ego, nxt);
    post_kernel<<<postGrid, THREADS, 0, stream>>>(
        ego, nxt, acc, noise + (size_t)k * total,
        (k == N_LAYERS - 1) ? (1.0f / (float)N_LAYERS) : 0.0f);
  }
}